// Encoder_15324443312556
// MI455X (gfx1250) — compile-verified
//
#include <hip/hip_runtime.h>

// ---------------------------------------------------------------------------
// GraphSAGE 2-layer forward, MI455X (gfx1250, wave32).
//   agg = segment_mean(x[src] -> dst);  h = lrelu(agg@Wl^T + x@Wr^T + b, 0.5)
// Aggregation: wave-per-edge float4 atomics (L2-resident: feature mats are
// 51.2MB << 192MB L2). GEMM: V_WMMA_F32_16X16X4_F32, 32(M)x16(N) tile per
// wave = 4 independent WMMA accumulation chains; deg-inv scaling folded into
// the epilogue (keeps WMMA inputs straight off the loads).
// ---------------------------------------------------------------------------

#define N_NODES 100000
#define N_EDGES 1600000
#define DIM     128
#define NS      0.5f

typedef float v2f __attribute__((ext_vector_type(2)));
typedef float v8f __attribute__((ext_vector_type(8)));

// ---------------------------------------------------------------- zero fill
__global__ void k_zero(float* __restrict__ p, int n4) {
    int i = blockIdx.x * blockDim.x + threadIdx.x;
    int stride = gridDim.x * blockDim.x;
    float4 z = make_float4(0.f, 0.f, 0.f, 0.f);
    for (; i < n4; i += stride)
        reinterpret_cast<float4*>(p)[i] = z;
}

// ------------------------------------------------------------- degree count
__global__ void k_deg_count(const int* __restrict__ dst, float* __restrict__ deg) {
    int e = blockIdx.x * blockDim.x + threadIdx.x;
    if (e < N_EDGES)
        unsafeAtomicAdd(&deg[dst[e]], 1.0f);
}

// ------------------------------------------------------- deg -> 1/max(deg,1)
__global__ void k_deg_inv(float* __restrict__ deg) {
    int i = blockIdx.x * blockDim.x + threadIdx.x;
    if (i < N_NODES)
        deg[i] = 1.0f / fmaxf(deg[i], 1.0f);
}

// ------------------------------------------------- scatter-add: agg[dst]+=f[src]
// one wave per edge; lane moves float4 (wave = 512B coalesced, hits L2)
__global__ __launch_bounds__(256)
void k_scatter(const float* __restrict__ feat,
               const int* __restrict__ src, const int* __restrict__ dst,
               float* __restrict__ agg) {
    int tid  = blockIdx.x * blockDim.x + threadIdx.x;
    int e    = tid >> 5;          // wave-uniform edge id
    int lane = tid & 31;          // 32 lanes x float4 = 128 floats
    if (e >= N_EDGES) return;
    int s = src[e], d = dst[e];
    const float4 v = reinterpret_cast<const float4*>(feat + (size_t)s * DIM)[lane];
    float* o = agg + (size_t)d * DIM + lane * 4;
    unsafeAtomicAdd(o + 0, v.x);
    unsafeAtomicAdd(o + 1, v.y);
    unsafeAtomicAdd(o + 2, v.z);
    unsafeAtomicAdd(o + 3, v.w);
}

// --------------------------------------------------------------- fused GEMM
// out[m][n] = lrelu( deginv[m]*sum_k agg[m][k]*Wl[n][k]
//                  + sum_k h[m][k]*Wr[n][k] + bias[n], 0.5 )
// Wave owns a 32(M)x16(N) tile (two 16-row sub-tiles -> B frags reused in
// registers, 4 independent WMMA chains). Block = 8 waves covers all 128
// cols of one 32-row stripe; grid = N/32 = 3125 blocks. K=128 in steps of 4
// via v_wmma_f32_16x16x4_f32.
//
// A frag (16x4 f32, ISA 7.12.2): lane<16 holds {K=k0,k0+1}, lane>=16 holds
// {K=k0+2,k0+3}, row M = lane&15  -> contiguous float2 load.
// B frag (4x16): symmetric, per-lane column n = lane&15, B[k][n] = W[n][k]
// -> contiguous float2 load from row-major W.
// C/D frag: VGPR v -> row v + 8*(lane>=16), col lane&15.
__global__ __launch_bounds__(256)
void k_sage_gemm(const float* __restrict__ agg, const float* __restrict__ hin,
                 const float* __restrict__ Wl,  const float* __restrict__ Wr,
                 const float* __restrict__ bias, const float* __restrict__ deginv,
                 float* __restrict__ out) {
    const int lane    = threadIdx.x & 31;
    const int wave    = threadIdx.x >> 5;        // 0..7 -> column tile
    const int rowBase = blockIdx.x * 32;
    const int half    = lane >> 4;               // K half select
    const int l15     = lane & 15;
    const int koff    = half * 2;
    const int col     = wave * 16 + l15;         // B/D column for this lane

    const float* a0 = agg + (size_t)(rowBase + l15)      * DIM + koff;
    const float* a1 = agg + (size_t)(rowBase + 16 + l15) * DIM + koff;
    const float* x0 = hin + (size_t)(rowBase + l15)      * DIM + koff;
    const float* x1 = hin + (size_t)(rowBase + 16 + l15) * DIM + koff;
    const float* wl = Wl  + (size_t)col * DIM + koff;
    const float* wr = Wr  + (size_t)col * DIM + koff;

    v8f accA0 = {0.f,0.f,0.f,0.f,0.f,0.f,0.f,0.f};
    v8f accX0 = accA0, accA1 = accA0, accX1 = accA0;

#pragma unroll 8
    for (int k0 = 0; k0 < DIM; k0 += 4) {
        v2f bl  = *reinterpret_cast<const v2f*>(wl + k0);   // shared across subtiles
        v2f br  = *reinterpret_cast<const v2f*>(wr + k0);
        v2f ag0 = *reinterpret_cast<const v2f*>(a0 + k0);
        v2f ax0 = *reinterpret_cast<const v2f*>(x0 + k0);
        v2f ag1 = *reinterpret_cast<const v2f*>(a1 + k0);
        v2f ax1 = *reinterpret_cast<const v2f*>(x1 + k0);
        // 4 independent accumulation chains
        accX0 = __builtin_amdgcn_wmma_f32_16x16x4_f32(false, ax0, false, br,
                                                      (short)0, accX0, false, false);
        accA0 = __builtin_amdgcn_wmma_f32_16x16x4_f32(false, ag0, false, bl,
                                                      (short)0, accA0, false, false);
        accX1 = __builtin_amdgcn_wmma_f32_16x16x4_f32(false, ax1, false, br,
                                                      (short)0, accX1, false, false);
        accA1 = __builtin_amdgcn_wmma_f32_16x16x4_f32(false, ag1, false, bl,
                                                      (short)0, accA1, false, false);
    }

    const float bc = bias[col];

    // epilogue: per-row deg-inv scale on the agg GEMM, bias, leaky_relu.
    // D rows for sub-tile t: m = rowBase + t*16 + half*8 + v, v = 0..7
#pragma unroll
    for (int t = 0; t < 2; ++t) {
        const int mBase = rowBase + t * 16 + half * 8;
        const float4 q0 = *reinterpret_cast<const float4*>(deginv + mBase);
        const float4 q1 = *reinterpret_cast<const float4*>(deginv + mBase + 4);
        const float dv[8] = {q0.x, q0.y, q0.z, q0.w, q1.x, q1.y, q1.z, q1.w};
        const v8f& aA = t ? accA1 : accA0;
        const v8f& aX = t ? accX1 : accX0;
#pragma unroll
        for (int v = 0; v < 8; ++v) {
            float y = fmaf(dv[v], aA[v], aX[v]) + bc;
            y = (y > 0.f) ? y : NS * y;
            out[(size_t)(mBase + v) * DIM + col] = y;
        }
    }
}

// ---------------------------------------------------------------------------
extern "C" void kernel_launch(void* const* d_in, const int* in_sizes, int n_in,
                              void* d_out, int out_size, void* d_ws, size_t ws_size,
                              hipStream_t stream) {
    const float* x   = (const float*)d_in[0];
    const int*   ei  = (const int*)d_in[1];       // [2, E]: src row then dst row
    const float* W1l = (const float*)d_in[2];
    const float* W1r = (const float*)d_in[3];
    const float* b1  = (const float*)d_in[4];
    const float* W2l = (const float*)d_in[5];
    const float* W2r = (const float*)d_in[6];
    const float* b2  = (const float*)d_in[7];
    float* out = (float*)d_out;

    const int* src = ei;
    const int* dst = ei + N_EDGES;

    // workspace: deg_inv (padded) | agg | h
    const size_t NPAD = 100096;                    // N rounded up, 16B aligned
    float* deg = (float*)d_ws;
    float* agg = deg + NPAD;
    float* h   = agg + (size_t)N_NODES * DIM;

    const int B = 256;
    const int zero1 = (int)((NPAD + (size_t)N_NODES * DIM) / 4);   // deg+agg, /float4
    const int zero2 = (int)(((size_t)N_NODES * DIM) / 4);          // agg only

    // deg + agg := 0
    k_zero<<<1024, B, 0, stream>>>(deg, zero1);
    // deg[dst] += 1
    k_deg_count<<<N_EDGES / B, B, 0, stream>>>(dst, deg);
    // deg := 1/max(deg,1)
    k_deg_inv<<<(N_NODES + B - 1) / B, B, 0, stream>>>(deg);
    // agg += x[src]  (wave per edge -> E*32 threads)
    k_scatter<<<N_EDGES / 8, B, 0, stream>>>(x, src, dst, agg);
    // h = lrelu(deginv*(agg @ W1l^T) + x @ W1r^T + b1)
    k_sage_gemm<<<N_NODES / 32, B, 0, stream>>>(agg, x, W1l, W1r, b1, deg, h);
    // agg := 0 ; agg += h[src]
    k_zero<<<1024, B, 0, stream>>>(agg, zero2);
    k_scatter<<<N_EDGES / 8, B, 0, stream>>>(h, src, dst, agg);
    // out = lrelu(deginv*(agg @ W2l^T) + h @ W2r^T + b2)
    k_sage_gemm<<<N_NODES / 32, B, 0, stream>>>(agg, h, W2l, W2r, b2, deg, out);
}